// Sparsify_all_74775380623608
// MI455X (gfx1250) — compile-verified
//
#include <hip/hip_runtime.h>
#include <cstdint>
#include <cstddef>

// Problem geometry (from reference): h is (64, 512, 32, 32) fp32.
#define ROWS      64
#define ROW_ELEMS 524288            // 512*32*32
#define KSEL      52429u            // k0+1, k0 = int(0.1*524288) = 52428 -> cutoff = 52429-th largest |h|
#define NBINS     65536             // 16-bit radix digit
#define SPLITS    8                 // blocks per row in histogram passes
#define TPB       1024              // 32 waves (wave32) per workgroup
#define STAGE_U32 4096              // one staging buffer: TPB lanes x 16B = 16 KB = 4096 u32
#define LDS_BYTES ((NBINS + 2 * STAGE_U32) * 4)   // 256KB bins + 32KB double buffer = 288KB

typedef __attribute__((ext_vector_type(4))) unsigned v4u;
typedef __attribute__((ext_vector_type(4))) float    v4f;

__device__ __forceinline__ unsigned absbits(unsigned u) { return u & 0x7FFFFFFFu; }

// gfx1250 async global->LDS copy, 16B per lane (ASYNCcnt-tracked, GVS mode).
__device__ __forceinline__ void async_ld16(unsigned lds_dst_bytes, unsigned global_off_bytes,
                                           const void* sbase) {
  asm volatile("global_load_async_to_lds_b128 %0, %1, %2"
               :
               : "v"(lds_dst_bytes), "v"(global_off_bytes), "s"(sbase)
               : "memory");
}
__device__ __forceinline__ void wait_async_le1() {
  asm volatile("s_wait_asynccnt 0x1" ::: "memory");
}
__device__ __forceinline__ void wait_async_0() {
  asm volatile("s_wait_asynccnt 0x0" ::: "memory");
}

// ---------------------------------------------------------------------------
// Zero the per-row global histograms (64 x 65536 u32 = 16 MB, scratch in d_out).
// ---------------------------------------------------------------------------
__global__ void zero_hist_k(unsigned* __restrict__ ghist) {
  size_t i = (size_t)blockIdx.x * blockDim.x + threadIdx.x;   // 1024*1024 uint4
  ((uint4*)ghist)[i] = make_uint4(0u, 0u, 0u, 0u);
}

// ---------------------------------------------------------------------------
// Pass 1: per-block LDS histogram of top-16 bits of |h| bit patterns.
// Data is staged global->LDS with double-buffered async-to-LDS b128 copies
// (no load-return VGPRs; tile K+1 in flight while tile K is histogrammed).
// 288 KB dynamic LDS per workgroup (CDNA5 WGP: 320 KB).
// ---------------------------------------------------------------------------
__global__ void hist_hi_k(const float* __restrict__ h, unsigned* __restrict__ ghist) {
  extern __shared__ unsigned bins[];                 // [0, NBINS): bins; then 2 staging bufs
  const int row = blockIdx.y, split = blockIdx.x, tid = threadIdx.x;

  for (int i = tid; i < NBINS; i += TPB) bins[i] = 0u;
  __syncthreads();

  unsigned* stage = bins + NBINS;
  // Wave-relative LDS byte address of the staging area (generic addr low 32 bits).
  const unsigned stage_lds = (unsigned)(size_t)(const void*)stage;
  const char* gbase = (const char*)(h + (size_t)row * ROW_ELEMS
                                      + (size_t)split * (ROW_ELEMS / SPLITS));

  // Prologue: tile 0 into buffer 0.
  async_ld16(stage_lds + tid * 16u, (unsigned)tid * 16u, gbase);

  #pragma unroll
  for (int k = 0; k < 16; ++k) {                     // 16 tiles x 16KB = 256KB per block
    const int buf = k & 1;
    if (k + 1 < 16) {
      const int nbuf = (k + 1) & 1;
      async_ld16(stage_lds + (unsigned)nbuf * 16384u + tid * 16u,
                 (unsigned)(tid + (k + 1) * TPB) * 16u, gbase);
      wait_async_le1();                              // oldest (tile k) has landed in LDS
    } else {
      wait_async_0();                                // drain tail
    }
    uint4 v = *(const uint4*)(stage + (size_t)buf * STAGE_U32 + (size_t)tid * 4);
    atomicAdd(&bins[absbits(v.x) >> 16], 1u);        // ds_add; forces dscnt wait on the
    atomicAdd(&bins[absbits(v.y) >> 16], 1u);        // ds_load before next overwrite
    atomicAdd(&bins[absbits(v.z) >> 16], 1u);
    atomicAdd(&bins[absbits(v.w) >> 16], 1u);
  }
  __syncthreads();

  unsigned* gh = ghist + (size_t)row * NBINS;
  for (int i = tid; i < NBINS; i += TPB) {
    unsigned c = bins[i];
    if (c) atomicAdd(&gh[i], c);                     // only populated bins (~4K of 64K)
  }
}

// ---------------------------------------------------------------------------
// Select the high 16-bit bin containing the KSEL-th largest element.
// meta[row] = bin b; meta[64+row] = remaining rank K' within bin b.
// Then re-zero this row's histogram for reuse by pass 2.
// ---------------------------------------------------------------------------
__global__ void select_hi_k(unsigned* __restrict__ ghist, unsigned* __restrict__ meta) {
  __shared__ unsigned csum[TPB];
  const int row = blockIdx.x, tid = threadIdx.x;
  unsigned* gh = ghist + (size_t)row * NBINS;

  unsigned s = 0;
  #pragma unroll 8
  for (int i = 0; i < 64; ++i) s += gh[tid * 64 + i];
  csum[tid] = s;
  __syncthreads();

  if (tid == 0) {
    unsigned cum = 0;
    int c = TPB - 1;
    for (; c > 0; --c) {                              // descending chunk scan
      if (cum + csum[c] >= KSEL) break;
      cum += csum[c];
    }
    unsigned b = 0, kk = 1;
    for (int bin = c * 64 + 63; bin >= c * 64; --bin) {
      unsigned cnt = gh[bin];
      if (cum + cnt >= KSEL) { b = (unsigned)bin; kk = KSEL - cum; break; }
      cum += cnt;
    }
    meta[row]      = b;
    meta[64 + row] = kk;
  }
  __syncthreads();
  for (int i = tid; i < NBINS; i += TPB) gh[i] = 0u;  // reuse for low-16 histogram
}

// ---------------------------------------------------------------------------
// Pass 2: for elements whose top-16 bits == b(row), histogram low 16 bits.
// Only ~1.4K elements per row match -> direct global atomics, no LDS needed.
// This re-read of h should be largely L2-resident (134MB + 16MB < 192MB L2).
// ---------------------------------------------------------------------------
__global__ void hist_lo_k(const float* __restrict__ h, unsigned* __restrict__ ghist,
                          const unsigned* __restrict__ meta) {
  const int row = blockIdx.y, split = blockIdx.x, tid = threadIdx.x;
  const unsigned b = meta[row];
  unsigned* gh = ghist + (size_t)row * NBINS;

  const uint4* p = (const uint4*)(h + (size_t)row * ROW_ELEMS)
                   + (size_t)split * (ROW_ELEMS / SPLITS / 4);
  #pragma unroll 4
  for (int k = 0; k < 16; ++k) {
    if (k + 2 < 16) __builtin_prefetch(&p[tid + (k + 2) * TPB], 0, 0);  // global_prefetch_b8
    uint4 v = p[tid + k * TPB];
    unsigned a;
    a = absbits(v.x); if ((a >> 16) == b) atomicAdd(&gh[a & 0xFFFFu], 1u);
    a = absbits(v.y); if ((a >> 16) == b) atomicAdd(&gh[a & 0xFFFFu], 1u);
    a = absbits(v.z); if ((a >> 16) == b) atomicAdd(&gh[a & 0xFFFFu], 1u);
    a = absbits(v.w); if ((a >> 16) == b) atomicAdd(&gh[a & 0xFFFFu], 1u);
  }
}

// ---------------------------------------------------------------------------
// Select low 16 bits -> exact cutoff bit pattern: meta[128+row] = (b<<16)|l.
// ---------------------------------------------------------------------------
__global__ void select_lo_k(const unsigned* __restrict__ ghist, unsigned* __restrict__ meta) {
  __shared__ unsigned csum[TPB];
  const int row = blockIdx.x, tid = threadIdx.x;
  const unsigned* gh = ghist + (size_t)row * NBINS;
  const unsigned kk = meta[64 + row];

  unsigned s = 0;
  #pragma unroll 8
  for (int i = 0; i < 64; ++i) s += gh[tid * 64 + i];
  csum[tid] = s;
  __syncthreads();

  if (tid == 0) {
    unsigned cum = 0;
    int c = TPB - 1;
    for (; c > 0; --c) {
      if (cum + csum[c] >= kk) break;
      cum += csum[c];
    }
    unsigned l = 0;
    for (int bin = c * 64 + 63; bin >= c * 64; --bin) {
      unsigned cnt = gh[bin];
      if (cum + cnt >= kk) { l = (unsigned)bin; break; }
      cum += cnt;
    }
    meta[128 + row] = (meta[row] << 16) | l;          // exact cutoff bit pattern
  }
}

// ---------------------------------------------------------------------------
// Final elementwise pass: out = |h| >= cutoff ? h : h*(1-tau).
// (mask*h*t + h*(1-t) == h when mask, h*(1-t) otherwise)
// Non-temporal load/store: write-once output must not thrash L2.
// ---------------------------------------------------------------------------
__global__ void apply_k(const float* __restrict__ h, const float* __restrict__ tau,
                        const unsigned* __restrict__ meta, float* __restrict__ out) {
  const int row = blockIdx.y;
  const unsigned cb = meta[128 + row];
  const float scale = 1.0f - tau[0];

  const v4u* p = (const v4u*)h;
  v4f* o = (v4f*)out;
  const size_t base4 = (size_t)row * (ROW_ELEMS / 4) + (size_t)blockIdx.x * 512;

  #pragma unroll
  for (int k = 0; k < 2; ++k) {
    size_t idx = base4 + threadIdx.x + (size_t)k * 256;
    v4u v = __builtin_nontemporal_load(&p[idx]);
    v4f r;
    r.x = (absbits(v.x) >= cb) ? __uint_as_float(v.x) : __uint_as_float(v.x) * scale;
    r.y = (absbits(v.y) >= cb) ? __uint_as_float(v.y) : __uint_as_float(v.y) * scale;
    r.z = (absbits(v.z) >= cb) ? __uint_as_float(v.z) : __uint_as_float(v.z) * scale;
    r.w = (absbits(v.w) >= cb) ? __uint_as_float(v.w) : __uint_as_float(v.w) * scale;
    __builtin_nontemporal_store(r, &o[idx]);
  }
}

// ---------------------------------------------------------------------------
extern "C" void kernel_launch(void* const* d_in, const int* in_sizes, int n_in,
                              void* d_out, int out_size, void* d_ws, size_t ws_size,
                              hipStream_t stream) {
  const float* h   = (const float*)d_in[0];
  const float* tau = (const float*)d_in[1];
  float* out       = (float*)d_out;

  // Scratch layout:
  //  - 64 x 65536 u32 histograms (16 MB) live in d_out (fully overwritten by apply_k).
  //  - meta in d_ws: [0..63]=high bin b, [64..127]=residual rank K', [128..191]=cutoff bits.
  unsigned* ghist = (unsigned*)d_out;
  unsigned* meta  = (unsigned*)d_ws;

  // Allow 288 KB dynamic LDS for the histogram kernel (CDNA5 WGP: 320 KB).
  (void)hipFuncSetAttribute(reinterpret_cast<const void*>(hist_hi_k),
                            hipFuncAttributeMaxDynamicSharedMemorySize,
                            (int)LDS_BYTES);

  zero_hist_k<<<(ROWS * NBINS / 4) / TPB, TPB, 0, stream>>>(ghist);
  hist_hi_k  <<<dim3(SPLITS, ROWS), TPB, LDS_BYTES, stream>>>(h, ghist);
  select_hi_k<<<ROWS, TPB, 0, stream>>>(ghist, meta);
  hist_lo_k  <<<dim3(SPLITS, ROWS), TPB, 0, stream>>>(h, ghist, meta);
  select_lo_k<<<ROWS, TPB, 0, stream>>>(ghist, meta);
  apply_k    <<<dim3(ROW_ELEMS / 2048, ROWS), 256, 0, stream>>>(h, tau, meta, out);
}